// ModelWithLoop_91130616086857
// MI455X (gfx1250) — compile-verified
//
#include <hip/hip_runtime.h>
#include <hip/hip_bf16.h>
#include <stdint.h>

// Problem geometry (matches the reference)
#define ROWS        512
#define N_SEG       1024
#define SEG_LEN     256
#define CROP        192
#define TOTAL_COLS  (N_SEG * SEG_LEN)   // 262144 input columns
#define OUT_COLS    (N_SEG * CROP)      // 196608 output columns

// TDM tiling: each workgroup moves a 64-row x 192-col tile, split into
// 4 sub-tiles of 16 rows (12 KB each) so the Tensor Data Mover has four
// descriptors in flight per phase. Total LDS = 48 KB -> up to 6 concurrent
// workgroups per WGP (320 KB LDS), overlapping load phases with store phases.
#define TILE_ROWS      64
#define SUB_TILES      4
#define SUB_ROWS       (TILE_ROWS / SUB_TILES)          // 16
#define SUB_ELEMS      (SUB_ROWS * CROP)                // 3072 floats = 12 KB
#define TILES_PER_SEG  (ROWS / TILE_ROWS)               // 8

typedef unsigned int u32x4 __attribute__((ext_vector_type(4)));
typedef int          i32x4 __attribute__((ext_vector_type(4)));
typedef int          i32x8 __attribute__((ext_vector_type(8)));

// ---------------- Tensor DMA descriptor construction (CDNA5 D#) ----------------
// Group 0 (128b): [1:0]=count(1 valid), [63:32]=lds_addr, [120:64]=global_addr,
//                 [127:126]=type(2="image"). gather/restore bits = 0.
__device__ __forceinline__ u32x4 tdm_group0(uint32_t lds_byte_addr, uint64_t gaddr) {
    u32x4 g0;
    g0.x = 1u;                                      // count = 1 (valid descriptor)
    g0.y = lds_byte_addr;                           // LDS byte address of tile start
    g0.z = (uint32_t)(gaddr & 0xFFFFFFFFu);         // global_addr[31:0]
    g0.w = ((uint32_t)(gaddr >> 32) & 0x01FFFFFFu)  // global_addr[56:32]
         | 0x80000000u;                             // type = 2 -> bits[127:126] = 10b
    return g0;
}

// Group 1 (256b): [17:16]=data_size(2 -> 4B), [79:48]=tensor_dim0,
// [111:80]=tensor_dim1, [127:112]=tile_dim0, [143:128]=tile_dim1,
// [159:144]=tile_dim2(0), [207:160]=tensor_dim0_stride (48b, elements).
// workgroup_mask=0 (not in a cluster), no padding/iteration/barrier.
__device__ __forceinline__ i32x8 tdm_group1(uint32_t tensor_d0, uint32_t tensor_d1,
                                            uint32_t tile_d0,   uint32_t tile_d1,
                                            uint32_t d0_stride_elems) {
    i32x8 g1;
    g1[0] = (int)(2u << 16);                                     // data_size = 4 bytes
    g1[1] = (int)((tensor_d0 & 0xFFFFu) << 16);                  // tensor_dim0[15:0]
    g1[2] = (int)(((tensor_d0 >> 16) & 0xFFFFu)                  // tensor_dim0[31:16]
                 | ((tensor_d1 & 0xFFFFu) << 16));               // tensor_dim1[15:0]
    g1[3] = (int)(((tensor_d1 >> 16) & 0xFFFFu)                  // tensor_dim1[31:16]
                 | ((tile_d0 & 0xFFFFu) << 16));                 // tile_dim0
    g1[4] = (int)(tile_d1 & 0xFFFFu);                            // tile_dim1 (tile_dim2=0)
    g1[5] = (int)d0_stride_elems;                                // tensor_dim0_stride[31:0]
    g1[6] = 0;                                                   // stride[47:32]=0, dim1_stride lo=0
    g1[7] = 0;                                                   // dim1_stride hi=0 (2D tile)
    return g1;
}

__device__ __forceinline__ void tdm_load_to_lds(u32x4 g0, i32x8 g1) {
    i32x4 z4 = {0, 0, 0, 0};
#if __clang_major__ >= 23
    i32x8 z8 = {0, 0, 0, 0, 0, 0, 0, 0};
    __builtin_amdgcn_tensor_load_to_lds(g0, g1, z4, z4, z8, 0);
#else
    __builtin_amdgcn_tensor_load_to_lds(g0, g1, z4, z4, 0);
#endif
}

__device__ __forceinline__ void tdm_store_from_lds(u32x4 g0, i32x8 g1) {
    i32x4 z4 = {0, 0, 0, 0};
#if __clang_major__ >= 23
    i32x8 z8 = {0, 0, 0, 0, 0, 0, 0, 0};
    __builtin_amdgcn_tensor_store_from_lds(g0, g1, z4, z4, z8, 0);
#else
    __builtin_amdgcn_tensor_store_from_lds(g0, g1, z4, z4, 0);
#endif
}

// ------------------------------- Kernel ---------------------------------------
// One single-wave workgroup per (segment, 64-row chunk):
//   issue 4x TDM loads  W[row0+16i : row0+16(i+1), start:start+192] -> LDS
//   s_wait_tensorcnt 0            (all loads landed; no assumption on L/S order)
//   issue 4x TDM stores LDS -> out[..., seg*192 : seg*192+192]
// The wave only builds descriptors; the Tensor Data Mover streams the bytes.
// A generic clamped-gather fallback keeps it correct for segments < 192 wide
// (never taken with the reference splits).
__global__ __launch_bounds__(32) void crop_copy_tdm(const float* __restrict__ W,
                                                    const int*   __restrict__ splits,
                                                    float*       __restrict__ out) {
    __shared__ __align__(16) float lds_tile[SUB_TILES * SUB_ELEMS];   // 48 KB

    const int bid   = blockIdx.x;
    const int seg   = bid / TILES_PER_SEG;
    const int chunk = bid % TILES_PER_SEG;
    const int row0  = chunk * TILE_ROWS;

    const long start = (long)splits[seg];
    const long end   = (long)splits[seg + 1];

    if (__builtin_expect(end - start < (long)CROP, 0)) {
        // Generic clamped gather (reference semantics): col = min(start+off, end-1).
        const long last = end - 1;
#pragma clang loop unroll(disable)
        for (int rr = 0; rr < TILE_ROWS; ++rr) {
#pragma clang loop unroll(disable)
            for (int cc = (int)threadIdx.x; cc < CROP; cc += 32) {
                long col = start + (long)cc;
                if (col > last) col = last;
                out[(size_t)(row0 + rr) * OUT_COLS + (size_t)seg * CROP + (size_t)cc] =
                    W[(size_t)(row0 + rr) * TOTAL_COLS + (size_t)col];
            }
        }
        return;
    }

    // Fast path: the crop is a contiguous 768B slice per row -> pure 2D tile DMA.
    const uint64_t ga_in   = (uint64_t)(uintptr_t)(W   + (size_t)row0 * TOTAL_COLS + (size_t)start);
    const uint64_t ga_out  = (uint64_t)(uintptr_t)(out + (size_t)row0 * OUT_COLS  + (size_t)seg * CROP);
    // Low 32 bits of the flat shared-pointer are the LDS byte offset.
    const uint32_t lds_addr = (uint32_t)(uintptr_t)(&lds_tile[0]);

    // Phase 1: four 16-row sub-tile loads in flight.
#pragma unroll
    for (int i = 0; i < SUB_TILES; ++i) {
        tdm_load_to_lds(
            tdm_group0(lds_addr + (uint32_t)i * (SUB_ELEMS * 4u),
                       ga_in + (uint64_t)i * (uint64_t)SUB_ROWS * TOTAL_COLS * 4ull),
            tdm_group1(/*tensor_d0=*/CROP, /*tensor_d1=*/SUB_ROWS,
                       /*tile_d0=*/CROP,   /*tile_d1=*/SUB_ROWS,
                       /*stride=*/TOTAL_COLS));
    }

    __builtin_amdgcn_s_wait_tensorcnt(0);   // all loads complete -> LDS tile valid

    // Phase 2: four sub-tile stores in flight (drained by s_wait_tensorcnt before exit).
#pragma unroll
    for (int i = 0; i < SUB_TILES; ++i) {
        tdm_store_from_lds(
            tdm_group0(lds_addr + (uint32_t)i * (SUB_ELEMS * 4u),
                       ga_out + (uint64_t)i * (uint64_t)SUB_ROWS * OUT_COLS * 4ull),
            tdm_group1(/*tensor_d0=*/CROP, /*tensor_d1=*/SUB_ROWS,
                       /*tile_d0=*/CROP,   /*tile_d1=*/SUB_ROWS,
                       /*stride=*/OUT_COLS));
    }

    __builtin_amdgcn_s_wait_tensorcnt(0);   // stores drained before wave exit
}

extern "C" void kernel_launch(void* const* d_in, const int* in_sizes, int n_in,
                              void* d_out, int out_size, void* d_ws, size_t ws_size,
                              hipStream_t stream) {
    const float* W      = (const float*)d_in[0];
    const int*   splits = (const int*)d_in[1];
    float*       out    = (float*)d_out;

    const dim3 grid(N_SEG * TILES_PER_SEG);   // 8192 single-wave workgroups
    const dim3 block(32);                     // one wave32 issues the DMAs
    crop_copy_tdm<<<grid, block, 0, stream>>>(W, splits, out);
}